// LogitDistance_28484223107945
// MI455X (gfx1250) — compile-verified
//
#include <hip/hip_runtime.h>
#include <hip/hip_bf16.h>

typedef __attribute__((ext_vector_type(2))) float v2f;
typedef __attribute__((ext_vector_type(8))) float v8f;

#define BATCH 16
#define NDIM  4096
#define TPB   256

// Reduce one wave32's per-lane partial sums with V_WMMA_F32_16X16X4_F32:
// A[m, K=0] = acc(lane m), A[m, K=2] = acc(lane m+16), other K zero; B = ones.
// D[m, n] = acc[m] + acc[m+16] (replicated across columns n).
// Per the 16x16 f32 C/D layout, lane L<16 holds D[k, L] in d[k] (rows 0..7),
// lane L>=16 holds rows 8..15. Folding d[0..7] locally gives: lanes 0-15 all
// hold Σ rows0-7, lanes 16-31 all hold Σ rows8-15; lane0 + lane16 = wave total.
__device__ __forceinline__ float wave_reduce_wmma(float acc, float* lds_acc) {
    v2f a;  a[0] = acc;  a[1] = 0.0f;
    v2f b;  b[0] = 1.0f; b[1] = 1.0f;
    v8f c = {};
    v8f d = __builtin_amdgcn_wmma_f32_16x16x4_f32(
        /*neg_a=*/false, a, /*neg_b=*/false, b,
        /*c_mod=*/(short)0, c, /*reuse_a=*/false, /*reuse_b=*/false);
    float s = ((d[0] + d[1]) + (d[2] + d[3])) + ((d[4] + d[5]) + (d[6] + d[7]));
    const int lane = threadIdx.x & 31;
    if (lane == 0 || lane == 16) atomicAdd(lds_acc, s);   // ds_add_f32
    return s;
}

__global__ void __launch_bounds__(TPB)
logitdist_pair_kernel(const float* __restrict__ pred, float* __restrict__ ws) {
    __shared__ float row[NDIM];
    __shared__ float blockAcc;
    const int b     = blockIdx.x >> 4;   // 16 i-chunks per batch row
    const int chunk = blockIdx.x & 15;
    const int tid   = threadIdx.x;

    if (tid == 0) blockAcc = 0.0f;

    // Stage the whole batch row into LDS once (16 KB): 4 x float4 per thread.
    const float4* src  = (const float4*)(pred + (size_t)b * NDIM);
    float4*       dstv = (float4*)row;
#pragma unroll
    for (int k = 0; k < 4; ++k)
        dstv[tid + k * TPB] = src[tid + k * TPB];
    __syncthreads();

    // Each thread owns one i; stream all j from LDS as broadcast b128 loads.
    const float xi = row[chunk * TPB + tid];
    float acc = 0.0f;
#pragma unroll 4
    for (int j = 0; j < NDIM; j += 4) {
        const float4 r = *(const float4*)&row[j];
        acc += fabsf(xi - r.x) + fabsf(xi - r.y)
             + fabsf(xi - r.z) + fabsf(xi - r.w);
    }

    // Wave32 reduction on the matrix pipe, then block combine in LDS.
    wave_reduce_wmma(acc, &blockAcc);
    __syncthreads();
    if (tid == 0) ws[blockIdx.x] = blockAcc;   // every slot written every call
}

__global__ void __launch_bounds__(TPB)
logitdist_final_kernel(const float* __restrict__ ws, float* __restrict__ out) {
    __shared__ float blockAcc;
    const int tid = threadIdx.x;
    if (tid == 0) blockAcc = 0.0f;
    __syncthreads();

    float acc = ws[tid];                  // exactly 256 block sums
    wave_reduce_wmma(acc, &blockAcc);
    __syncthreads();

    if (tid == 0) {
        // out = (S/2) / (ntriu * B * N), S = full ordered pairwise |diff| sum
        const double ntriu = 0.5 * (double)NDIM * (double)(NDIM - 1);
        const double scale = 0.5 / (ntriu * (double)BATCH * (double)NDIM);
        out[0] = (float)((double)blockAcc * scale);
    }
}

extern "C" void kernel_launch(void* const* d_in, const int* in_sizes, int n_in,
                              void* d_out, int out_size, void* d_ws, size_t ws_size,
                              hipStream_t stream) {
    (void)in_sizes; (void)n_in; (void)out_size; (void)ws_size;
    const float* pred = (const float*)d_in[0];
    float*       ws   = (float*)d_ws;    // 256 floats of scratch
    float*       out  = (float*)d_out;

    // 16 batches x 16 i-chunks = 256 blocks of 256 threads (8 wave32s each).
    logitdist_pair_kernel<<<dim3(BATCH * (NDIM / TPB)), dim3(TPB), 0, stream>>>(pred, ws);
    logitdist_final_kernel<<<dim3(1), dim3(TPB), 0, stream>>>(ws, out);
}